// SpatiotemporalAttention_29927332118815
// MI455X (gfx1250) — compile-verified
//
#include <hip/hip_runtime.h>
#include <hip/hip_bf16.h>

typedef __attribute__((ext_vector_type(2))) float v2f;
typedef __attribute__((ext_vector_type(4))) float v4f;
typedef __attribute__((ext_vector_type(8))) float v8f;

// GCC-style int4 for the async-to-LDS builtin's pointer params
typedef int v4i_gcc __attribute__((vector_size(16)));
typedef __attribute__((address_space(1))) v4i_gcc* gv4i_p;   // global
typedef __attribute__((address_space(3))) v4i_gcc* lv4i_p;   // LDS

constexpr int Bn = 2;     // batch
constexpr int Cn = 256;   // channels
constexpr int Nn = 4096;  // D*H*W

// ---------------------------------------------------------------------------
// Kernel 1: gex[b,n] = sum_c g[c]*ex[b,c,n], gq likewise.
// ---------------------------------------------------------------------------
__global__ void __launch_bounds__(256)
gvec_kernel(const float* __restrict__ ex, const float* __restrict__ q,
            const float* __restrict__ wg, float* __restrict__ gex,
            float* __restrict__ gq) {
  int g = blockIdx.x * 256 + threadIdx.x;        // 0 .. 2*Bn*Nn-1
  int sel = g / (Bn * Nn);
  int rem = g - sel * (Bn * Nn);
  int b = rem / Nn;
  int n = rem - b * Nn;
  const float* src = sel ? q : ex;
  const float* col = src + (size_t)b * Cn * Nn + n;
  float acc = 0.f;
  for (int c = 0; c < Cn; ++c) acc += wg[c] * col[(size_t)c * Nn];
  (sel ? gq : gex)[rem] = acc;
}

// ---------------------------------------------------------------------------
// Kernel 2: corr projections.  out[b,n,d] = sum_c X[b,c,n] * W[d,c]
// One wave32 per 16x16 output tile, V_WMMA_F32_16X16X4_F32, K-loop of 64.
// f32 fragment layouts per CDNA5 ISA 7.12.2:
//   A 16x4 : lane(0-15)=M,  vgpr0 K=2*hi, vgpr1 K=2*hi+1
//   B 4x16 : lane(0-15)=N,  vgpr0 K=2*hi, vgpr1 K=2*hi+1
//   D 16x16: vgpr r -> row r+8*hi, col = lane%16
// ---------------------------------------------------------------------------
__global__ void __launch_bounds__(256)
proj_kernel(const float* __restrict__ ex, const float* __restrict__ q,
            const float* __restrict__ Wex, const float* __restrict__ Wq,
            float* __restrict__ excorr, float* __restrict__ qcorr) {
  int wave = blockIdx.x * 8 + (threadIdx.x >> 5);
  int lane = threadIdx.x & 31;
  int lo = lane & 15, hi = lane >> 4;

  int cblk = wave & 15;  wave >>= 4;      // Cn/16 = 16
  int nblk = wave & 255; wave >>= 8;      // Nn/16 = 256
  int b    = wave & 1;
  int sel  = wave >> 1;

  const float* X  = sel ? q : ex;
  const float* Wm = sel ? Wq : Wex;
  float* out      = sel ? qcorr : excorr;

  int n0 = nblk * 16, c0 = cblk * 16;
  const float* xcol = X + (size_t)b * Cn * Nn + n0 + lo;   // A[m=lo][k] = xcol[k*Nn]
  const float* wrow = Wm + (size_t)(c0 + lo) * Cn;         // B[k][j=lo] = wrow[k]

  v8f acc;
  for (int i = 0; i < 8; ++i) acc[i] = 0.f;

  for (int kk = 0; kk < Cn; kk += 4) {
    int kA = kk + 2 * hi;
    v2f a, bfrag;
    a.x     = xcol[(size_t)kA * Nn];
    a.y     = xcol[(size_t)(kA + 1) * Nn];
    bfrag.x = wrow[kA];
    bfrag.y = wrow[kA + 1];
    acc = __builtin_amdgcn_wmma_f32_16x16x4_f32(false, a, false, bfrag,
                                                (short)0, acc, false, false);
  }

  float* orow = out + (size_t)(b * Nn + n0 + 8 * hi) * Cn + c0 + lo;
  for (int r = 0; r < 8; ++r) orow[(size_t)r * Cn] = acc[r];
}

// ---------------------------------------------------------------------------
// Kernel 3: fused scores + softmax + mask + transposed-A writeback.
// One workgroup (8 waves) per (sel, b, 32-row block of corr).  Each wave
// computes TWO 16x16 tiles per 128-col chunk, reusing its B fragments.
//   Phase 1: online row max/sumexp/gdot over 32 chunks of 128 cols.
//   Phase 2: recompute scores, write P^T = A output, normalized.
// ---------------------------------------------------------------------------
__global__ void __launch_bounds__(256)
attn_kernel(const float* __restrict__ ex, const float* __restrict__ q,
            const float* __restrict__ excorr, const float* __restrict__ qcorr,
            const float* __restrict__ gex, const float* __restrict__ gq,
            float* __restrict__ out) {
  constexpr int CP = 260;   // A-tile pitch: 16B-aligned rows, bank-conflict free
  constexpr int SP = 132;   // padded S-chunk row pitch
  __shared__ float Atile[32 * CP];
  __shared__ float Schunk[32 * SP];
  __shared__ float redm[256], reds[256], redd[256];
  __shared__ float rmax[32], rsum[32], rdot[32];

  int blk  = blockIdx.x;                 // 0..511
  int nblk = blk & 127; blk >>= 7;       // Nn/32 = 128
  int b    = blk & 1;
  int sel  = blk >> 1;

  const float* Arows = (sel ? qcorr : excorr) + (size_t)(b * Nn + nblk * 32) * Cn;
  const float* Bsrc  = (sel ? ex : q) + (size_t)b * Cn * Nn;   // [c][m]
  const float* gv    = (sel ? gq : gex) + (size_t)b * Nn;
  float* maskout = out + (size_t)sel * (Bn * Nn) + (size_t)b * Nn;
  float* Aout    = out + (size_t)2 * Bn * Nn
                       + (size_t)sel * Bn * Nn * Nn + (size_t)b * Nn * Nn;

  int tid = threadIdx.x;
  // ---- Stage the 32x256 A-tile into LDS (async DMA path when available) ----
#if __has_builtin(__builtin_amdgcn_global_load_async_to_lds_b128)
  for (int i = tid; i < 32 * Cn / 4; i += 256) {       // 2048 16B chunks
    int r = i >> 6;             // 64 chunks per row
    int c = (i & 63) * 4;
    __builtin_amdgcn_global_load_async_to_lds_b128(
        (gv4i_p)(uintptr_t)(Arows + (size_t)r * Cn + c),
        (lv4i_p)(unsigned)(uintptr_t)&Atile[r * CP + c],
        0, 0);
  }
#if __has_builtin(__builtin_amdgcn_s_wait_asynccnt)
  __builtin_amdgcn_s_wait_asynccnt(0);
#else
  asm volatile("s_wait_asynccnt 0" ::: "memory");
#endif
#else
  for (int i = tid; i < 32 * Cn; i += 256) {
    int r = i >> 8, c = i & 255;
    Atile[r * CP + c] = Arows[(size_t)r * Cn + c];
  }
#endif
  if (tid < 32) { rmax[tid] = -3.0e38f; rsum[tid] = 0.f; rdot[tid] = 0.f; }
  __syncthreads();

  int lane = tid & 31, w = tid >> 5;
  int lo = lane & 15, hi = lane >> 4;
  int row32 = tid >> 3, seg = tid & 7;   // reduction: 8 threads/row, 16 cols each

  // ---------------- Phase 1: softmax statistics + mask dot ----------------
  for (int ch = 0; ch < 32; ++ch) {
    int m0 = ch * 128 + w * 16;
    const float* bcol = Bsrc + m0 + lo;
    __builtin_prefetch(bcol + 128, 0, 1);   // global_prefetch_b8: next chunk
    v8f acc0, acc1;
    for (int i = 0; i < 8; ++i) { acc0[i] = 0.f; acc1[i] = 0.f; }
    for (int kk = 0; kk < Cn; kk += 4) {
      int kA = kk + 2 * hi;
      v2f a0, a1, bfrag;
      a0.x = Atile[lo * CP + kA];        a0.y = Atile[lo * CP + kA + 1];
      a1.x = Atile[(16 + lo) * CP + kA]; a1.y = Atile[(16 + lo) * CP + kA + 1];
      bfrag.x = bcol[(size_t)kA * Nn];
      bfrag.y = bcol[(size_t)(kA + 1) * Nn];
      acc0 = __builtin_amdgcn_wmma_f32_16x16x4_f32(false, a0, false, bfrag,
                                                   (short)0, acc0, false, false);
      acc1 = __builtin_amdgcn_wmma_f32_16x16x4_f32(false, a1, false, bfrag,
                                                   (short)0, acc1, false, false);
    }
    for (int r = 0; r < 8; ++r) {
      Schunk[(r + 8 * hi) * SP + w * 16 + lo]      = acc0[r];
      Schunk[(16 + r + 8 * hi) * SP + w * 16 + lo] = acc1[r];
    }
    __syncthreads();

    // Thread (row32, seg) reduces cols [seg*16, seg*16+16) of its row.
    int base = row32 * SP + seg * 16;
    float lmax = -3.0e38f;
    for (int i = 0; i < 16; ++i) lmax = fmaxf(lmax, Schunk[base + i]);
    redm[row32 * 8 + seg] = lmax;
    __syncthreads();

    float nm = rmax[row32];
    for (int i = 0; i < 8; ++i) nm = fmaxf(nm, redm[row32 * 8 + i]);
    float lsum = 0.f, ldot = 0.f;
    int cbase = ch * 128 + seg * 16;
    for (int i = 0; i < 16; ++i) {
      float e = __expf(Schunk[base + i] - nm);
      lsum += e;
      ldot += e * gv[cbase + i];
    }
    reds[row32 * 8 + seg] = lsum;
    redd[row32 * 8 + seg] = ldot;
    __syncthreads();

    if (seg == 0) {
      float s = 0.f, dd = 0.f;
      for (int i = 0; i < 8; ++i) { s += reds[row32 * 8 + i]; dd += redd[row32 * 8 + i]; }
      float scale = __expf(rmax[row32] - nm);
      rsum[row32] = rsum[row32] * scale + s;
      rdot[row32] = rdot[row32] * scale + dd;
      rmax[row32] = nm;
    }
    __syncthreads();
  }

  // mask[b, row] = sigmoid( sum_k P[row,k]*gv[k] ) = sigmoid(rdot/rsum)
  if (tid < 32) {
    float x = rdot[tid] / rsum[tid];
    maskout[nblk * 32 + tid] = 1.f / (1.f + __expf(-x));
  }
  __syncthreads();

  // ---------------- Phase 2: recompute scores, write P^T ------------------
  for (int ch = 0; ch < 32; ++ch) {
    int m0 = ch * 128 + w * 16;
    const float* bcol = Bsrc + m0 + lo;
    __builtin_prefetch(bcol + 128, 0, 1);
    v8f acc0, acc1;
    for (int i = 0; i < 8; ++i) { acc0[i] = 0.f; acc1[i] = 0.f; }
    for (int kk = 0; kk < Cn; kk += 4) {
      int kA = kk + 2 * hi;
      v2f a0, a1, bfrag;
      a0.x = Atile[lo * CP + kA];        a0.y = Atile[lo * CP + kA + 1];
      a1.x = Atile[(16 + lo) * CP + kA]; a1.y = Atile[(16 + lo) * CP + kA + 1];
      bfrag.x = bcol[(size_t)kA * Nn];
      bfrag.y = bcol[(size_t)(kA + 1) * Nn];
      acc0 = __builtin_amdgcn_wmma_f32_16x16x4_f32(false, a0, false, bfrag,
                                                   (short)0, acc0, false, false);
      acc1 = __builtin_amdgcn_wmma_f32_16x16x4_f32(false, a1, false, bfrag,
                                                   (short)0, acc1, false, false);
    }
    // Lane holds col = m0+lo; rows are contiguous in A's last (n) dim.
    float* dst = Aout + (size_t)(m0 + lo) * Nn + nblk * 32 + 8 * hi;
    v4f p0, p1, p2, p3;
    for (int r = 0; r < 4; ++r) {
      p0[r] = __expf(acc0[r]     - rmax[r + 8 * hi])          / rsum[r + 8 * hi];
      p1[r] = __expf(acc0[r + 4] - rmax[r + 4 + 8 * hi])      / rsum[r + 4 + 8 * hi];
      p2[r] = __expf(acc1[r]     - rmax[16 + r + 8 * hi])     / rsum[16 + r + 8 * hi];
      p3[r] = __expf(acc1[r + 4] - rmax[16 + r + 4 + 8 * hi]) / rsum[16 + r + 4 + 8 * hi];
    }
    *(v4f*)(dst)      = p0;
    *(v4f*)(dst + 4)  = p1;
    *(v4f*)(dst + 16) = p2;
    *(v4f*)(dst + 20) = p3;
  }
}

// ---------------------------------------------------------------------------
extern "C" void kernel_launch(void* const* d_in, const int* in_sizes, int n_in,
                              void* d_out, int out_size, void* d_ws, size_t ws_size,
                              hipStream_t stream) {
  const float* ex  = (const float*)d_in[0];
  const float* q   = (const float*)d_in[1];
  const float* Wex = (const float*)d_in[2];
  const float* Wq  = (const float*)d_in[3];
  const float* wg  = (const float*)d_in[4];

  float* ws     = (float*)d_ws;
  float* excorr = ws;                                   // Bn*Nn*Cn
  float* qcorr  = excorr + (size_t)Bn * Nn * Cn;        // Bn*Nn*Cn
  float* gex    = qcorr + (size_t)Bn * Nn * Cn;         // Bn*Nn
  float* gq     = gex + (size_t)Bn * Nn;                // Bn*Nn
  float* out    = (float*)d_out;

  gvec_kernel<<<(2 * Bn * Nn) / 256, 256, 0, stream>>>(ex, q, wg, gex, gq);
  // tiles = 2(sel) * Bn * (Nn/16) * (Cn/16) = 16384 waves, 8 waves/block
  proj_kernel<<<2048, 256, 0, stream>>>(ex, q, Wex, Wq, excorr, qcorr);
  // blocks = 2(sel) * Bn * (Nn/32) = 512
  attn_kernel<<<512, 256, 0, stream>>>(ex, q, excorr, qcorr, gex, gq, out);
}